// TTTMetaAdapter_9088150798893
// MI455X (gfx1250) — compile-verified
//
#include <hip/hip_runtime.h>
#include <math.h>
#include <stdint.h>

// Problem constants (from reference setup_inputs)
constexpr int BB   = 2;
constexpr int SS   = 2048;
constexpr int NR   = BB * SS;      // 4096 token rows
constexpr int HH   = 1024;
constexpr int RR   = 16;
constexpr int VV   = 32000;
constexpr int NTOK = BB * (SS - 1); // 4094
constexpr float LR = 0.01f;
constexpr int VB   = VV / 128;     // 250 vocab tiles

typedef __attribute__((ext_vector_type(2))) float v2f;
typedef __attribute__((ext_vector_type(8))) float v8f;

// Tensor-DMA availability (device pass only; host pass takes fallback)
#if defined(__AMDGCN__) && __has_builtin(__builtin_amdgcn_tensor_load_to_lds) && \
    __has_builtin(__builtin_amdgcn_s_wait_tensorcnt)
#define USE_TDM 1
#else
#define USE_TDM 0
#endif

__device__ __forceinline__ v8f zero8() {
    v8f z;
#pragma unroll
    for (int i = 0; i < 8; ++i) z[i] = 0.0f;
    return z;
}

// D(16x16,f32) = A(16x4,f32) * B(4x16,f32) + C   — V_WMMA_F32_16X16X4_F32
__device__ __forceinline__ v8f wmma4(v2f a, v2f b, v8f c) {
    return __builtin_amdgcn_wmma_f32_16x16x4_f32(false, a, false, b, (short)0, c,
                                                 false, false);
}

#if USE_TDM
typedef __attribute__((ext_vector_type(4))) unsigned int tdm_v4u;
typedef __attribute__((ext_vector_type(8))) int tdm_v8i;
typedef __attribute__((ext_vector_type(4))) int tdm_v4i;

// Issue a TDM 2-D tile load Global->LDS (fp32 elements).
//   tile_d0 : elements per row (innermost, contiguous)
//   tile_d1 : number of rows
//   stride0 : row pitch of the source matrix, in elements
//   pad_int : LDS pad interval code  (code k => pad every 2^(k+1) DWORDs)
//   pad_amt : LDS pad amount code    (code k => k+1 DWORDs of pad)
// D# layout per CDNA5 ISA (group0 128b, group1 256b).
__device__ __forceinline__ void tdm_load_2d(const void* gptr, unsigned lds_addr,
                                            unsigned tile_d0, unsigned tile_d1,
                                            unsigned long long stride0,
                                            unsigned pad_int, unsigned pad_amt)
{
    unsigned long long ga = (unsigned long long)(uintptr_t)gptr;

    tdm_v4u g0;
    g0[0] = 1u;                                   // count=1 valid descriptor
    g0[1] = lds_addr;                             // lds_addr  (bits 63:32)
    g0[2] = (unsigned)ga;                         // global_addr[31:0]
    g0[3] = (unsigned)((ga >> 32) & 0x01FFFFFFull)// global_addr[56:32]
          | (2u << 30);                           // type=2 ("image")

    tdm_v8i g1;
    g1[0] = (int)((2u << 16)                      // data_size = 4 bytes
                | (1u << 20)                      // pad_enable
                | (pad_int << 22)                 // pad_interval
                | (pad_amt << 25));               // pad_amount
    g1[1] = (int)((tile_d0 & 0xFFFFu) << 16);     // tensor_dim0[15:0] @bit48
    g1[2] = (int)(((tile_d0 >> 16) & 0xFFFFu)     // tensor_dim0[31:16]
                | ((tile_d1 & 0xFFFFu) << 16));   // tensor_dim1[15:0] @bit80
    g1[3] = (int)(((tile_d1 >> 16) & 0xFFFFu)     // tensor_dim1[31:16]
                | ((tile_d0 & 0xFFFFu) << 16));   // tile_dim0 @bit112
    g1[4] = (int)(tile_d1 & 0xFFFFu);             // tile_dim1 @bit128 (tile_dim2=0)
    g1[5] = (int)(stride0 & 0xFFFFFFFFull);       // tensor_dim0_stride[31:0]
    g1[6] = (int)((stride0 >> 32) & 0xFFFFull);   // tensor_dim0_stride[47:32]
    g1[7] = 0;                                    // tensor_dim1_stride (2-D)

    tdm_v4i gz; gz[0] = gz[1] = gz[2] = gz[3] = 0;
#if defined(__clang_major__) && (__clang_major__ >= 23)
    tdm_v8i gz8;
#pragma unroll
    for (int i = 0; i < 8; ++i) gz8[i] = 0;
    __builtin_amdgcn_tensor_load_to_lds(g0, g1, gz, gz, gz8, 0);
#else
    __builtin_amdgcn_tensor_load_to_lds(g0, g1, gz, gz, 0);
#endif
}
#define LDS_BYTE_ADDR(p) ((unsigned)(unsigned long long)(uintptr_t)(p))
#endif  // USE_TDM

// ---------------------------------------------------------------------------
// K1/K8: out = hs + (hs @ Am) @ Bm ; optionally store u = hs @ Am
// grid = NR/128 blocks, 256 threads (8 waves). Wave w owns row sub-tile w.
// ---------------------------------------------------------------------------
__global__ __launch_bounds__(256) void k_adapter(
    const float* __restrict__ hs,   // NR x HH
    const float* __restrict__ Am,   // HH x RR
    const float* __restrict__ Bm,   // RR x HH
    float* __restrict__ outp,       // NR x HH
    float* __restrict__ u_out)      // NR x RR (may be null)
{
    __shared__ float hs_s[128][36];  // 128 x 32 K-tile of hs (pad 36)
    __shared__ float a_s[32][17];    // a_s[k][r]
    __shared__ float u_s[128][17];   // u band
    __shared__ float b_s[16][132];   // Bm column chunk

    const int tid = threadIdx.x;
    const int wave = tid >> 5, lane = tid & 31, hi = lane >> 4, lo = lane & 15;
    const int n0 = blockIdx.x * 128;

    // ---- phase 1: u(128x16) = hs_band(128x1024) @ Am(1024x16)
    v8f cu = zero8();
    for (int kt = 0; kt < HH; kt += 32) {
        for (int i = tid; i < 128 * 32; i += 256) {
            int r = i >> 5, c = i & 31;
            hs_s[r][c] = hs[(size_t)(n0 + r) * HH + kt + c];
        }
        for (int i = tid; i < 32 * 16; i += 256) {
            int k = i >> 4, r = i & 15;
            a_s[k][r] = Am[(size_t)(kt + k) * RR + r];
        }
        __syncthreads();
#pragma unroll
        for (int kc = 0; kc < 32; kc += 4) {
            v2f a, b;
            a[0] = hs_s[wave * 16 + lo][kc + 2 * hi];
            a[1] = hs_s[wave * 16 + lo][kc + 2 * hi + 1];
            b[0] = a_s[kc + 2 * hi][lo];
            b[1] = a_s[kc + 2 * hi + 1][lo];
            cu = wmma4(a, b, cu);
        }
        __syncthreads();
    }
#pragma unroll
    for (int j = 0; j < 8; ++j) {
        int nl = wave * 16 + j + 8 * hi;
        u_s[nl][lo] = cu[j];
        if (u_out) u_out[(size_t)(n0 + nl) * RR + lo] = cu[j];
    }
    __syncthreads();

    // ---- phase 2: out = hs + u(128x16) @ Bm(16x1024), 128-col chunks
    for (int h0 = 0; h0 < HH; h0 += 128) {
        for (int i = tid; i < 16 * 128; i += 256) {
            int k = i >> 7, c = i & 127;
            b_s[k][c] = Bm[(size_t)k * HH + h0 + c];
        }
        __syncthreads();
        v8f cc[8];
#pragma unroll
        for (int i = 0; i < 8; ++i) cc[i] = zero8();
#pragma unroll
        for (int kc = 0; kc < 16; kc += 4) {
            v2f a;
            a[0] = u_s[wave * 16 + lo][kc + 2 * hi];
            a[1] = u_s[wave * 16 + lo][kc + 2 * hi + 1];
#pragma unroll
            for (int ns = 0; ns < 8; ++ns) {
                v2f b;
                b[0] = b_s[kc + 2 * hi][ns * 16 + lo];
                b[1] = b_s[kc + 2 * hi + 1][ns * 16 + lo];
                cc[ns] = wmma4(a, b, cc[ns]);
            }
        }
#pragma unroll
        for (int ns = 0; ns < 8; ++ns)
#pragma unroll
            for (int j = 0; j < 8; ++j) {
                size_t idx = (size_t)(n0 + wave * 16 + j + 8 * hi) * HH
                           + h0 + ns * 16 + lo;
                outp[idx] = hs[idx] + cc[ns][j];
            }
        __syncthreads();
    }
}

// ---------------------------------------------------------------------------
// K2: bW(16xV) = b(16x1024) @ W(1024xV).  grid = VB blocks.
// ---------------------------------------------------------------------------
__global__ __launch_bounds__(256) void k_bw(
    const float* __restrict__ bm, const float* __restrict__ W,
    float* __restrict__ bW)
{
    __shared__ float Ws[32][132];
    __shared__ float bs[32][17];
    const int tid = threadIdx.x;
    const int wave = tid >> 5, lane = tid & 31, hi = lane >> 4, lo = lane & 15;
    const int v0 = blockIdx.x * 128;

    v8f cc = zero8();
    for (int kt = 0; kt < HH; kt += 32) {
        for (int i = tid; i < 32 * 128; i += 256) {
            int k = i >> 7, c = i & 127;
            Ws[k][c] = W[(size_t)(kt + k) * VV + v0 + c];
        }
        for (int i = tid; i < 32 * 16; i += 256) {
            int k = i >> 4, r = i & 15;
            bs[k][r] = bm[(size_t)r * HH + kt + k];
        }
        __syncthreads();
#pragma unroll
        for (int kc = 0; kc < 32; kc += 4) {
            v2f a, b;
            a[0] = bs[kc + 2 * hi][lo];
            a[1] = bs[kc + 2 * hi + 1][lo];
            b[0] = Ws[kc + 2 * hi][wave * 16 + lo];
            b[1] = Ws[kc + 2 * hi + 1][wave * 16 + lo];
            cc = wmma4(a, b, cc);
        }
        __syncthreads();
    }
#pragma unroll
    for (int j = 0; j < 8; ++j)
        bW[(size_t)(j + 8 * hi) * VV + v0 + wave * 16 + lo] = cc[j];
}

// ---------------------------------------------------------------------------
// K3: logits = adapted @ W.  grid = (VB, NR/128). MT=NT=128, KT=16.
// 268 GFLOP f32 — the dominant kernel.  TDM path: wave 0 drives a
// double-buffered tensor_load_to_lds pipeline (TENSORcnt) while all 8 waves
// issue WMMAs; LDS padding (pitch 20 / 132) comes from the D# pad fields.
// ---------------------------------------------------------------------------
__global__ __launch_bounds__(256) void k_logits(
    const float* __restrict__ Aa,   // NR x HH (adapted)
    const float* __restrict__ W,    // HH x VV
    float* __restrict__ logits)     // NR x VV
{
    const int tid = threadIdx.x;
    const int wave = tid >> 5, lane = tid & 31, hi = lane >> 4, lo = lane & 15;
    const int v0 = blockIdx.x * 128;
    const int m0 = blockIdx.y * 128;

    v8f cc[8];
#pragma unroll
    for (int i = 0; i < 8; ++i) cc[i] = zero8();

#if USE_TDM
    __shared__ float As[2][128][20];   // A tile: 128 rows x 16 k, pitch 20
    __shared__ float Bs[2][16][132];   // B tile: 16 k x 128 v, pitch 132

    // Prologue: stage tile 0 into buffer 0.
    if (wave == 0) {
        tdm_load_2d(&Aa[(size_t)m0 * HH], LDS_BYTE_ADDR(&As[0][0][0]),
                    /*d0=*/16, /*d1=*/128, /*stride=*/HH, /*pad_int=*/3, /*pad_amt=*/3);
        tdm_load_2d(&W[(size_t)0 * VV + v0], LDS_BYTE_ADDR(&Bs[0][0][0]),
                    /*d0=*/128, /*d1=*/16, /*stride=*/VV, /*pad_int=*/6, /*pad_amt=*/3);
    }
    const int NT = HH / 16;  // 64 K-tiles
    for (int t = 0; t < NT; ++t) {
        const int buf = t & 1;
        if (wave == 0) {
            if (t + 1 < NT) {  // issue next tile, then wait for current (in-order)
                const int kt = (t + 1) * 16;
                tdm_load_2d(&Aa[(size_t)m0 * HH + kt], LDS_BYTE_ADDR(&As[buf ^ 1][0][0]),
                            16, 128, HH, 3, 3);
                tdm_load_2d(&W[(size_t)kt * VV + v0], LDS_BYTE_ADDR(&Bs[buf ^ 1][0][0]),
                            128, 16, VV, 6, 3);
                __builtin_amdgcn_s_wait_tensorcnt(2);
            } else {
                __builtin_amdgcn_s_wait_tensorcnt(0);
            }
        }
        __syncthreads();  // tile t resident in LDS for all waves
#pragma unroll
        for (int kc = 0; kc < 16; kc += 4) {
            v2f a;
            a[0] = As[buf][wave * 16 + lo][kc + 2 * hi];
            a[1] = As[buf][wave * 16 + lo][kc + 2 * hi + 1];
#pragma unroll
            for (int ns = 0; ns < 8; ++ns) {
                v2f b;
                b[0] = Bs[buf][kc + 2 * hi][ns * 16 + lo];
                b[1] = Bs[buf][kc + 2 * hi + 1][ns * 16 + lo];
                cc[ns] = wmma4(a, b, cc[ns]);
            }
        }
        __syncthreads();  // done reading buf before TDM overwrites it (t+2)
    }
#else
    __shared__ float As[128][20];
    __shared__ float Bs[16][132];
    for (int kt = 0; kt < HH; kt += 16) {
        for (int i = tid; i < 128 * 16; i += 256) {
            int r = i >> 4, c = i & 15;
            As[r][c] = Aa[(size_t)(m0 + r) * HH + kt + c];
        }
        for (int i = tid; i < 16 * 128; i += 256) {
            int k = i >> 7, c = i & 127;
            Bs[k][c] = W[(size_t)(kt + k) * VV + v0 + c];
        }
        if (kt + 16 < HH) {  // prefetch next W tile (global_prefetch_b8)
            __builtin_prefetch(W + (size_t)(kt + 16 + (tid >> 7)) * VV + v0 + (tid & 127), 0, 1);
        }
        __syncthreads();
#pragma unroll
        for (int kc = 0; kc < 16; kc += 4) {
            v2f a;
            a[0] = As[wave * 16 + lo][kc + 2 * hi];
            a[1] = As[wave * 16 + lo][kc + 2 * hi + 1];
#pragma unroll
            for (int ns = 0; ns < 8; ++ns) {
                v2f b;
                b[0] = Bs[kc + 2 * hi][ns * 16 + lo];
                b[1] = Bs[kc + 2 * hi + 1][ns * 16 + lo];
                cc[ns] = wmma4(a, b, cc[ns]);
            }
        }
        __syncthreads();
    }
#endif
#pragma unroll
    for (int ns = 0; ns < 8; ++ns)
#pragma unroll
        for (int j = 0; j < 8; ++j)
            logits[(size_t)(m0 + wave * 16 + j + 8 * hi) * VV + v0 + ns * 16 + lo]
                = cc[ns][j];
}

// ---------------------------------------------------------------------------
// K4: per-row softmax stats + per-row NLL (deterministic, no atomics).
// grid = NR blocks, one row each.
// ---------------------------------------------------------------------------
__global__ __launch_bounds__(256) void k_stats(
    const float* __restrict__ logits, const int* __restrict__ ids,
    float* __restrict__ m_ws, float* __restrict__ is_ws,
    int* __restrict__ tgt_ws, float* __restrict__ nll)
{
    __shared__ float red[256];
    const int n = blockIdx.x, tid = threadIdx.x;
    const float* row = logits + (size_t)n * VV;

    float mx = -3.4e38f;
    for (int v = tid; v < VV; v += 256) mx = fmaxf(mx, row[v]);
    red[tid] = mx; __syncthreads();
    for (int s = 128; s > 0; s >>= 1) {
        if (tid < s) red[tid] = fmaxf(red[tid], red[tid + s]);
        __syncthreads();
    }
    mx = red[0]; __syncthreads();

    float se = 0.0f;
    for (int v = tid; v < VV; v += 256) se += __expf(row[v] - mx);
    red[tid] = se; __syncthreads();
    for (int s = 128; s > 0; s >>= 1) {
        if (tid < s) red[tid] += red[tid + s];
        __syncthreads();
    }
    se = red[0];

    if (tid == 0) {
        m_ws[n]  = mx;
        is_ws[n] = 1.0f / se;
        int b = n / SS, sp = n % SS;
        int tgt = -1;
        float e = 0.0f;
        if (sp < SS - 1) {
            tgt = ids[b * SS + sp + 1];
            e = mx + __logf(se) - row[tgt];
        }
        tgt_ws[n] = tgt;
        nll[n] = e;
    }
}

// ---------------------------------------------------------------------------
// K5a: per 128-vocab slab, loop all row bands:
//   P = (softmax - onehot)/NTOK ; T_slab = u^T @ P ; q_part = P @ bW_slab^T
// grid = VB blocks. T is exact (slab-owned); q_part reduced later.
// ---------------------------------------------------------------------------
__global__ __launch_bounds__(256) void k_pass2(
    const float* __restrict__ logits, const float* __restrict__ u,
    const float* __restrict__ bW,
    const float* __restrict__ m_ws, const float* __restrict__ is_ws,
    const int* __restrict__ tgt_ws,
    float* __restrict__ T,          // RR x VV
    float* __restrict__ q_part)     // VB x NR x RR
{
    __shared__ float P_s[128][132];
    __shared__ float bW_s[16][132];
    __shared__ float u_s[128][17];
    __shared__ float mrow[128], crow[128];
    __shared__ int   trow[128];

    const int tid = threadIdx.x;
    const int wave = tid >> 5, lane = tid & 31, hi = lane >> 4, lo = lane & 15;
    const int v0 = blockIdx.x * 128;
    const float invN = 1.0f / (float)NTOK;

    for (int i = tid; i < 16 * 128; i += 256) {
        int r = i >> 7, c = i & 127;
        bW_s[r][c] = bW[(size_t)r * VV + v0 + c];
    }

    v8f cT = zero8();
    for (int nb = 0; nb < NR; nb += 128) {
        for (int i = tid; i < 128 * 16; i += 256) {
            int k = i >> 4, r = i & 15;
            u_s[k][r] = u[(size_t)(nb + k) * RR + r];
        }
        if (tid < 128) {
            int t = tgt_ws[nb + tid];
            trow[tid] = t;
            mrow[tid] = m_ws[nb + tid];
            crow[tid] = (t >= 0) ? is_ws[nb + tid] * invN : 0.0f;
        }
        __syncthreads();

        for (int i = tid; i < 128 * 128; i += 256) {
            int nl = i >> 7, vl = i & 127;
            float l = logits[(size_t)(nb + nl) * VV + v0 + vl];
            float p = __expf(l - mrow[nl]) * crow[nl];
            if (v0 + vl == trow[nl]) p -= invN;
            P_s[nl][vl] = p;
        }
        __syncthreads();

        // T += u_band^T (16 x 128n) @ P (128n x 128v); wave w -> v sub-tile w
#pragma unroll 4
        for (int kc = 0; kc < 128; kc += 4) {
            v2f a, b;
            a[0] = u_s[kc + 2 * hi][lo];
            a[1] = u_s[kc + 2 * hi + 1][lo];
            b[0] = P_s[kc + 2 * hi][wave * 16 + lo];
            b[1] = P_s[kc + 2 * hi + 1][wave * 16 + lo];
            cT = wmma4(a, b, cT);
        }
        // q_band = P (128n x 128v) @ bW_slab^T (128v x 16); wave w -> n sub-tile w
        v8f cq = zero8();
#pragma unroll 4
        for (int kc = 0; kc < 128; kc += 4) {
            v2f a, b;
            a[0] = P_s[wave * 16 + lo][kc + 2 * hi];
            a[1] = P_s[wave * 16 + lo][kc + 2 * hi + 1];
            b[0] = bW_s[lo][kc + 2 * hi];
            b[1] = bW_s[lo][kc + 2 * hi + 1];
            cq = wmma4(a, b, cq);
        }
#pragma unroll
        for (int j = 0; j < 8; ++j) {
            int n = nb + wave * 16 + j + 8 * hi;
            q_part[((size_t)blockIdx.x * NR + n) * RR + lo] = cq[j];
        }
        __syncthreads();
    }
#pragma unroll
    for (int j = 0; j < 8; ++j)
        T[(size_t)(j + 8 * hi) * VV + v0 + wave * 16 + lo] = cT[j];
}

// ---------------------------------------------------------------------------
// K5b: gb(16 x H) = T(16 x V) @ W^T.  grid = HH/128 blocks. LDS-transposed W.
// ---------------------------------------------------------------------------
__global__ __launch_bounds__(256) void k_gb(
    const float* __restrict__ T, const float* __restrict__ W,
    float* __restrict__ gb)
{
    __shared__ float T_s[16][132];
    __shared__ float Wt_s[128][133];
    const int tid = threadIdx.x;
    const int wave = tid >> 5, lane = tid & 31, hi = lane >> 4, lo = lane & 15;
    const int h0 = blockIdx.x * 128;

    v8f cc = zero8();
    for (int vk = 0; vk < VV; vk += 128) {
        for (int i = tid; i < 16 * 128; i += 256) {
            int r = i >> 7, c = i & 127;
            T_s[r][c] = T[(size_t)r * VV + vk + c];
        }
        for (int i = tid; i < 128 * 128; i += 256) {
            int h = i >> 7, c = i & 127;
            Wt_s[c][h] = W[(size_t)(h0 + h) * VV + vk + c];  // transpose in LDS
        }
        __syncthreads();
#pragma unroll 4
        for (int kc = 0; kc < 128; kc += 4) {
            v2f a, b;
            a[0] = T_s[lo][kc + 2 * hi];
            a[1] = T_s[lo][kc + 2 * hi + 1];
            b[0] = Wt_s[kc + 2 * hi][wave * 16 + lo];
            b[1] = Wt_s[kc + 2 * hi + 1][wave * 16 + lo];
            cc = wmma4(a, b, cc);
        }
        __syncthreads();
    }
#pragma unroll
    for (int j = 0; j < 8; ++j)
        gb[(size_t)(j + 8 * hi) * HH + h0 + wave * 16 + lo] = cc[j];
}

// ---------------------------------------------------------------------------
// K5c: q = sum over vocab slabs of q_part.  grid = NR*RR/256 blocks.
// ---------------------------------------------------------------------------
__global__ __launch_bounds__(256) void k_qreduce(
    const float* __restrict__ q_part, float* __restrict__ q)
{
    int idx = blockIdx.x * 256 + threadIdx.x;  // < NR*RR
    float s = 0.0f;
    for (int vb = 0; vb < VB; ++vb)
        s += q_part[(size_t)vb * NR * RR + idx];
    q[idx] = s;
}

// ---------------------------------------------------------------------------
// K6: ga(H x 16) = hs^T @ q.  grid = HH/128 blocks. LDS-transposed hs.
// ---------------------------------------------------------------------------
__global__ __launch_bounds__(256) void k_ga(
    const float* __restrict__ hs, const float* __restrict__ q,
    float* __restrict__ ga)
{
    __shared__ float hsT_s[128][133];
    __shared__ float q_s[128][17];
    const int tid = threadIdx.x;
    const int wave = tid >> 5, lane = tid & 31, hi = lane >> 4, lo = lane & 15;
    const int h0 = blockIdx.x * 128;

    v8f cc = zero8();
    for (int nk = 0; nk < NR; nk += 128) {
        for (int i = tid; i < 128 * 128; i += 256) {
            int r = i >> 7, c = i & 127;
            hsT_s[c][r] = hs[(size_t)(nk + r) * HH + h0 + c];  // transpose in LDS
        }
        for (int i = tid; i < 128 * 16; i += 256) {
            int k = i >> 4, r = i & 15;
            q_s[k][r] = q[(size_t)(nk + k) * RR + r];
        }
        __syncthreads();
#pragma unroll 4
        for (int kc = 0; kc < 128; kc += 4) {
            v2f a, b;
            a[0] = hsT_s[wave * 16 + lo][kc + 2 * hi];
            a[1] = hsT_s[wave * 16 + lo][kc + 2 * hi + 1];
            b[0] = q_s[kc + 2 * hi][lo];
            b[1] = q_s[kc + 2 * hi + 1][lo];
            cc = wmma4(a, b, cc);
        }
        __syncthreads();
    }
#pragma unroll
    for (int j = 0; j < 8; ++j)
        ga[(size_t)(h0 + wave * 16 + j + 8 * hi) * RR + lo] = cc[j];
}

// ---------------------------------------------------------------------------
// K7: new_a/new_b, loss, grad_norm (single block, deterministic reductions)
// ---------------------------------------------------------------------------
__global__ __launch_bounds__(256) void k_finalize(
    const float* __restrict__ fa, const float* __restrict__ fb,
    const float* __restrict__ ga, const float* __restrict__ gb,
    const float* __restrict__ nll,
    float* __restrict__ out_a, float* __restrict__ out_b,
    float* __restrict__ out_loss, float* __restrict__ out_gn)
{
    __shared__ float red[256];
    const int tid = threadIdx.x;

    float ss = 0.0f;
    for (int i = tid; i < HH * RR; i += 256) {
        float g = ga[i]; ss += g * g; out_a[i] = fa[i] - LR * g;
    }
    for (int i = tid; i < RR * HH; i += 256) {
        float g = gb[i]; ss += g * g; out_b[i] = fb[i] - LR * g;
    }
    red[tid] = ss; __syncthreads();
    for (int s = 128; s > 0; s >>= 1) {
        if (tid < s) red[tid] += red[tid + s];
        __syncthreads();
    }
    if (tid == 0) out_gn[0] = sqrtf(red[0]);
    __syncthreads();

    float sl = 0.0f;
    for (int i = tid; i < NR; i += 256) sl += nll[i];
    red[tid] = sl; __syncthreads();
    for (int s = 128; s > 0; s >>= 1) {
        if (tid < s) red[tid] += red[tid + s];
        __syncthreads();
    }
    if (tid == 0) out_loss[0] = red[0] * (1.0f / (float)NTOK);
}

// ---------------------------------------------------------------------------
// Host-side launcher.  Workspace requirement ~612 MB (logits cache 524 MB +
// q partials 65.5 MB + adapted 16 MB + small buffers).
// ---------------------------------------------------------------------------
extern "C" void kernel_launch(void* const* d_in, const int* in_sizes, int n_in,
                              void* d_out, int out_size, void* d_ws, size_t ws_size,
                              hipStream_t stream)
{
    (void)in_sizes; (void)n_in; (void)out_size; (void)ws_size;

    const float* hs  = (const float*)d_in[0];   // (2,2048,1024)
    const int*   ids = (const int*)  d_in[1];   // (2,2048)
    const float* fa  = (const float*)d_in[2];   // (1024,16)
    const float* fb  = (const float*)d_in[3];   // (16,1024)
    const float* W   = (const float*)d_in[4];   // (1024,32000)

    float* out      = (float*)d_out;
    float* out_out  = out;                                   // NR*HH
    float* out_a    = out + (size_t)NR * HH;                 // HH*RR
    float* out_b    = out_a + (size_t)HH * RR;               // RR*HH
    float* out_loss = out_b + (size_t)RR * HH;               // 1
    float* out_gn   = out_loss + 1;                          // 1

    float* ws = (float*)d_ws;
    size_t off = 0;
    float* u       = ws + off; off += (size_t)NR * RR;       // 65,536
    float* adapted = ws + off; off += (size_t)NR * HH;       // 4,194,304
    float* bW      = ws + off; off += (size_t)RR * VV;       // 512,000
    float* T       = ws + off; off += (size_t)RR * VV;       // 512,000
    float* m_ws    = ws + off; off += NR;
    float* is_ws   = ws + off; off += NR;
    int*   tgt_ws  = (int*)(ws + off); off += NR;
    float* nll     = ws + off; off += NR;
    float* gbuf    = ws + off; off += (size_t)RR * HH;       // gb
    float* gabuf   = ws + off; off += (size_t)HH * RR;       // ga
    float* q       = ws + off; off += (size_t)NR * RR;
    float* q_part  = ws + off; off += (size_t)VB * NR * RR;  // 16,384,000
    float* logits  = ws + off; off += (size_t)NR * VV;       // 131,072,000

    // Forward adapter: u = hs@a, adapted = hs + u@b
    k_adapter<<<NR / 128, 256, 0, stream>>>(hs, fa, fb, adapted, u);
    // bW = b @ lm_head (rank-16 projection of W)
    k_bw<<<VB, 256, 0, stream>>>(fb, W, bW);
    // Dominant GEMM: logits = adapted @ W  (cached in ws, TDM double-buffered)
    k_logits<<<dim3(VB, NR / 128), 256, 0, stream>>>(adapted, W, logits);
    // Row softmax stats + per-row NLL
    k_stats<<<NR, 256, 0, stream>>>(logits, ids, m_ws, is_ws, tgt_ws, nll);
    // Backward rank-16 contractions: T = u^T P ; q_part = P bW^T
    k_pass2<<<VB, 256, 0, stream>>>(logits, u, bW, m_ws, is_ws, tgt_ws, T, q_part);
    // gb = T @ W^T
    k_gb<<<HH / 128, 256, 0, stream>>>(T, W, gbuf);
    // q = sum_slabs q_part
    k_qreduce<<<(NR * RR) / 256, 256, 0, stream>>>(q_part, q);
    // ga = hs^T @ q
    k_ga<<<HH / 128, 256, 0, stream>>>(hs, q, gabuf);
    // new_a/new_b, loss, grad_norm
    k_finalize<<<1, 256, 0, stream>>>(fa, fb, gabuf, gbuf, nll,
                                      out_a, out_b, out_loss, out_gn);
    // out = hs + (hs@new_a)@new_b
    k_adapter<<<NR / 128, 256, 0, stream>>>(hs, out_a, out_b, out_out, nullptr);
}